// DetectPeaksCC_76012331205178
// MI455X (gfx1250) — compile-verified
//
#include <hip/hip_runtime.h>

#define NT     4096
#define NC     3
#define NXB    64
#define NB     32
#define CHUNK  512               // floats per chunk (2 KB per wave)
#define NITER  (NT / CHUNK)      // 8 chunks per row
#define NBUF   4                 // LDS buffers per wave (3 chunks in flight + current)
#define NGRID  201

__device__ __forceinline__ void merge_top2(float& s0, int& i0, float& s1,
                                           float os0, int oi0, float os1) {
  // Merge two (top1, top1-index, top2) triples; smaller index wins ties (top_k order).
  if (os0 > s0 || (os0 == s0 && oi0 < i0)) {
    s1 = fmaxf(s0, os1);
    s0 = os0;
    i0 = oi0;
  } else {
    s1 = fmaxf(s1, os0);
  }
}

__global__ __launch_bounds__(96) void
DetectPeaksCC_76012331205178_kernel(const float* __restrict__ xcorr,
                                    const int* __restrict__ nlag_p,
                                    float* __restrict__ out) {
  __shared__ __align__(16) float sbuf[NC][NBUF][CHUNK]; // 24 KB: 3 waves x 4 x 2KB
  __shared__ float sres[NC][3];                         // {weight, score0, idxf}

  const int lane = threadIdx.x & 31;
  const int ch   = threadIdx.x >> 5;   // wave id == channel (0..2)
  const int bx   = blockIdx.x;         // b*64 + x
  const float NEG_INF = -__builtin_inff();

  const float* __restrict__ row =
      xcorr + (((size_t)(bx >> 6) * NC + ch) * NXB + (bx & 63)) * NT;

  const unsigned lds0   = (unsigned)(uintptr_t)(&sbuf[ch][0][0]);
  const unsigned lane_b = (unsigned)lane * 16u;   // 16 B per lane per b128 issue
  const unsigned long long gbase = (unsigned long long)(uintptr_t)row;

  // Warm the row into L2 (128 cachelines of 128 B).
#pragma unroll
  for (int k = 0; k < 4; ++k) {
    __builtin_prefetch(row + (k * 32 + lane) * 32, 0, 1);
  }

  // Issue one 2 KB chunk = 4x global_load_async_to_lds_b128 (512 B per wave each).
  auto issue_chunk = [&](int chunk) {
    const unsigned bufb = lds0 + (unsigned)(chunk & (NBUF - 1)) * (CHUNK * 4u);
    const unsigned gb   = (unsigned)chunk * (CHUNK * 4u);
#pragma unroll
    for (int k = 0; k < 4; ++k) {
      unsigned vdst = bufb + (unsigned)k * 512u + lane_b;
      unsigned voff = gb   + (unsigned)k * 512u + lane_b;
      asm volatile("global_load_async_to_lds_b128 %0, %1, %2"
                   :: "v"(vdst), "v"(voff), "s"(gbase) : "memory");
    }
  };

  // Prime the pipeline: chunks 0,1,2 in flight.
  issue_chunk(0);
  issue_chunk(1);
  issue_chunk(2);

  float s0 = NEG_INF, s1 = NEG_INF;
  int   i0 = 0x7fffffff;
  float carry = NEG_INF;   // abs(x[t0-1]) for lane 0 of the next 128-block

#pragma unroll
  for (int it = 0; it < NITER; ++it) {
    // Keep 3 chunks ahead; waits guarantee chunks `it` AND `it+1` have landed
    // (async loads complete in order; <=8 leaves only it+2/it+3's 8 issues).
    if (it < NITER - 3) {
      issue_chunk(it + 3);
      asm volatile("s_wait_asynccnt 0x8" ::: "memory");
    } else if (it == NITER - 3) {
      asm volatile("s_wait_asynccnt 0x4" ::: "memory");
    } else {
      asm volatile("s_wait_asynccnt 0x0" ::: "memory");
    }

    const float* buf  = &sbuf[ch][it & (NBUF - 1)][0];
    const float* bufn = &sbuf[ch][(it + 1) & (NBUF - 1)][0];

#pragma unroll
    for (int s = 0; s < 4; ++s) {
      const int off = s * 128 + lane * 4;
      const int t0  = it * CHUNK + off;

      float4 v = *reinterpret_cast<const float4*>(buf + off);
      float e0 = fabsf(v.x), e1 = fabsf(v.y), e2 = fabsf(v.z), e3 = fabsf(v.w);

      // lane 31's next-neighbor: uniform LDS broadcast read of the next block's head
      float bnd;
      if (s < 3)               bnd = fabsf(buf[(s + 1) * 128]);
      else if (it < NITER - 1) bnd = fabsf(bufn[0]);
      else                     bnd = NEG_INF;   // end of row: -inf padding

      // neighbors via wave shuffles (already abs'd by the producing lane)
      float pm = __shfl_up(e3, 1, 32);    // lane L <- lane L-1's last element
      float pn = __shfl_down(e0, 1, 32);  // lane L <- lane L+1's first element
      if (lane == 0)  pm = carry;
      if (lane == 31) pn = bnd;

      // fused 3-wide local-max test: candidate = (e >= max(left,right)) ? e : 0
      float c0 = (e0 >= fmaxf(pm, e1)) ? e0 : 0.0f;
      float c1 = (e1 >= fmaxf(e0, e2)) ? e1 : 0.0f;
      float c2 = (e2 >= fmaxf(e1, e3)) ? e2 : 0.0f;
      float c3 = (e3 >= fmaxf(e2, pn)) ? e3 : 0.0f;

      // branch-free top-2: s1' = max(s1, min(c,s0)); s0' = max(s0, c);
      // i0 on strict '>' against the *old* s0 keeps first-occurrence order.
      i0 = (c0 > s0) ? t0 + 0 : i0;  s1 = fmaxf(s1, fminf(c0, s0));  s0 = fmaxf(s0, c0);
      i0 = (c1 > s0) ? t0 + 1 : i0;  s1 = fmaxf(s1, fminf(c1, s0));  s0 = fmaxf(s0, c1);
      i0 = (c2 > s0) ? t0 + 2 : i0;  s1 = fmaxf(s1, fminf(c2, s0));  s0 = fmaxf(s0, c2);
      i0 = (c3 > s0) ? t0 + 3 : i0;  s1 = fmaxf(s1, fminf(c3, s0));  s0 = fmaxf(s0, c3);

      carry = __shfl(e3, 31, 32);  // lane 31's last element -> next block's prev
    }
  }

  // ---------- wave-wide top-2 reduction (wave32 butterfly) ----------
  for (int m = 16; m >= 1; m >>= 1) {
    float os0 = __shfl_xor(s0, m, 32);
    float os1 = __shfl_xor(s1, m, 32);
    int   oi0 = __shfl_xor(i0, m, 32);
    merge_top2(s0, i0, s1, os0, oi0, os1);
  }

  // ---------- 3-point parabola, argmax on 201-point grid ----------
  const int im1 = (i0 > 0) ? i0 - 1 : 0;
  const int ip1 = (i0 < NT - 1) ? i0 + 1 : NT - 1;
  const float ym1 = fabsf(row[im1]);
  const float y0  = fabsf(row[i0]);
  const float yp1 = fabsf(row[ip1]);
  const float qa = 0.5f * (ym1 + yp1) - y0;
  const float qb = 0.5f * (yp1 - ym1);
  const float qc = y0;

  float bg = NEG_INF;
  int   bj = NGRID;
  for (int j = lane; j < NGRID; j += 32) {
    float xg = -1.0f + 0.01f * (float)j;
    float yg = (qa * xg + qb) * xg + qc;
    if (yg > bg) { bg = yg; bj = j; }   // strict '>': first occurrence (argmax)
  }
  for (int m = 16; m >= 1; m >>= 1) {
    float og = __shfl_xor(bg, m, 32);
    int   oj = __shfl_xor(bj, m, 32);
    if (og > bg || (og == bg && oj < bj)) { bg = og; bj = oj; }
  }
  const float sub    = -1.0f + 0.01f * (float)bj;
  const float score0 = (qa * sub + qb) * sub + qc;
  const float idxf   = (float)i0 + sub;
  const float weight = (0.1f + 3.0f * (s0 - s1)) * s0 * s0;

  if (lane == 0) {
    sres[ch][0] = weight;
    sres[ch][1] = score0;
    sres[ch][2] = idxf;
  }
  __syncthreads();

  // ---------- channel argmax + output (thread 0) ----------
  if (threadIdx.x == 0) {
    int cbest = 0;
    float wbest = sres[0][0];
    if (sres[1][0] > wbest) { wbest = sres[1][0]; cbest = 1; }
    if (sres[2][0] > wbest) { wbest = sres[2][0]; cbest = 2; }
    const float nlag      = (float)nlag_p[0];
    const float shift_idx = sres[cbest][2] - nlag;
    out[0 * (NB * NXB) + bx] = sres[cbest][1];       // cc_max
    out[1 * (NB * NXB) + bx] = wbest;                // cc_weight
    out[2 * (NB * NXB) + bx] = shift_idx / 100.0f;   // cc_dt
    out[3 * (NB * NXB) + bx] = shift_idx;            // cc_shift
  }
}

extern "C" void kernel_launch(void* const* d_in, const int* in_sizes, int n_in,
                              void* d_out, int out_size, void* d_ws, size_t ws_size,
                              hipStream_t stream) {
  (void)in_sizes; (void)n_in; (void)out_size; (void)d_ws; (void)ws_size;
  const float* xcorr  = (const float*)d_in[0];
  const int*   nlag_p = (const int*)d_in[1];
  float*       out    = (float*)d_out;

  dim3 grid(NB * NXB);   // 2048 blocks, one per (b, x)
  dim3 block(96);        // 3 wave32s, one per channel
  DetectPeaksCC_76012331205178_kernel<<<grid, block, 0, stream>>>(xcorr, nlag_p, out);
}